// BlittingStrokeModel_2233382994395
// MI455X (gfx1250) — compile-verified
//
#include <hip/hip_runtime.h>
#include <math.h>

// Problem constants (from reference)
#define Nn 4
#define Pp 16
#define HH 384
#define WW 384
#define KK 8
#define BHW 64

typedef float v2f __attribute__((ext_vector_type(2)));
typedef float v8f __attribute__((ext_vector_type(8)));

// Extract D[p][p] (p = 0..15) from the 16x16 f32 C/D WMMA layout:
//   VGPR j: lanes 0-15 hold (M=j,   N=lane),
//           lanes 16-31 hold (M=j+8, N=lane-16)
// => diag p<8  : VGPR p,   lane p
//    diag p>=8 : VGPR p-8, lane (p-8)+24
__device__ __forceinline__ float diag16(v8f c, int p) {
  float r = 0.0f;
#pragma unroll
  for (int j = 0; j < 8; ++j) {
    float e0 = __shfl(c[j], j, 32);
    float e1 = __shfl(c[j], j + 24, 32);
    r = (p == j) ? e0 : r;
    r = (p == (j + 8)) ? e1 : r;
  }
  return r;
}

// One wave. Cubic spline eval via V_WMMA_F32_16X16X4_F32 (K=4 == cubic order).
// A (16x4): row p = [1, f_p, f_p^2, f_p^3]  (per ISA: lanes0-15 K0/K1 in v0/v1,
//                                            lanes16-31 K2/K3 in v0/v1)
// B (4x16): col p' = coeffs of segment(p')   (mirror layout of A)
// D[p][p] = value of p's polynomial at its own offset -> diagonal.
__global__ __launch_bounds__(32) void spline_params_kernel(
    const float* __restrict__ knot_t, const float* __restrict__ A_,
    const float* __restrict__ B_, const float* __restrict__ C2_,
    const float* __restrict__ D3_, const float* __restrict__ ts,
    float* __restrict__ par) {
  const int lane = threadIdx.x;   // 0..31, EXEC all ones (WMMA requirement)
  const int p = lane & 15;
  const float t = ts[p];

  // searchsorted(knot_t, t, 'right') - 1, clamped to [0, K-2]
  int cnt = 0;
#pragma unroll
  for (int i = 0; i < KK; ++i) cnt += (knot_t[i] <= t) ? 1 : 0;
  int idx = cnt - 1;
  idx = idx < 0 ? 0 : (idx > (KK - 2) ? (KK - 2) : idx);
  const float f = t - knot_t[idx];
  const float f2 = f * f, f3 = f2 * f;
  const bool lo = (lane < 16);

  v2f Am;
  Am.x = lo ? 1.0f : f2;   // K0 | K2
  Am.y = lo ? f    : f3;   // K1 | K3

  for (int n = 0; n < Nn; ++n) {
    float qv[3], qdv[3];
#pragma unroll
    for (int dim = 0; dim < 3; ++dim) {
      const int ci = (n * (KK - 1) + idx) * 3 + dim;
      const float ca = A_[ci], cb = B_[ci], cc = C2_[ci], cd = D3_[ci];

      // q  = a + b f + (c/2) f^2 + (d/3) f^3
      v2f Bq;
      Bq.x = lo ? ca : (cc * 0.5f);
      Bq.y = lo ? cb : (cd * (1.0f / 3.0f));
      // qd = b + c f + d f^2 (+ 0 f^3)
      v2f Bd;
      Bd.x = lo ? cb : cd;
      Bd.y = lo ? cc : 0.0f;

      v8f cq = {};
      cq = __builtin_amdgcn_wmma_f32_16x16x4_f32(false, Am, false, Bq,
                                                 (short)0, cq, false, false);
      v8f cdm = {};
      cdm = __builtin_amdgcn_wmma_f32_16x16x4_f32(false, Am, false, Bd,
                                                  (short)0, cdm, false, false);
      qv[dim] = diag16(cq, p);
      qdv[dim] = diag16(cdm, p);
    }
    const float x = qv[0], y = qv[1], z = qv[2];
    const float th = -atan2f(qdv[1], qdv[0]);
    float s = fminf(fmaxf(z, 0.001f), 1.0f);
    const float act = (z > 0.0f) ? 1.0f : 0.0f;
    if (lane < 16) {
      float* o = par + (n * Pp + p) * 8;
      o[0] = x;
      o[1] = y;
      o[2] = cosf(th);
      o[3] = sinf(th);
      o[4] = 1.0f / s;
      o[5] = act;
      o[6] = 0.0f;
      o[7] = 0.0f;
    }
  }
}

// One 32x32 tile per block, 4 pixels/thread (rows ty, ty+8, ty+16, ty+24).
// Brush alpha (64x64 f32, 16 KB) staged in LDS via CDNA5 async global->LDS
// b128 DMA; the 1-pixel zero pad of the reference is handled arithmetically
// with the same in-bounds indicator needed for the constant-color rgb mask.
__global__ __launch_bounds__(256) void render_kernel(
    const float* __restrict__ images, const float* __restrict__ colors,
    const float* __restrict__ brush, const float* __restrict__ par,
    float* __restrict__ out) {
  __shared__ __align__(16) float sA[BHW * BHW];  // 64x64 brush alpha
  __shared__ float sP[Pp * 8];                   // stroke params (uniform)
  __shared__ float sC[4];                        // colors[n]

  const int tid = threadIdx.x;
  const int n = blockIdx.z;

  // Async DMA: 4096 floats = 1024 x b128, 4 per thread. Rows are 256B so every
  // 16B chunk is aligned in both global and LDS.
  const float* __restrict__ galpha = brush + 3 * BHW * BHW;  // channel 3
#pragma unroll
  for (int k = 0; k < 4; ++k) {
    const int q4 = tid + k * 256;  // float4 index
    unsigned loff = (unsigned)(size_t)(&sA[q4 * 4]);  // low 32b = LDS offset
    unsigned long long gaddr = (unsigned long long)(size_t)(galpha + q4 * 4);
    asm volatile("global_load_async_to_lds_b128 %0, %1, off"
                 :
                 : "v"(loff), "v"(gaddr)
                 : "memory");
  }
  if (tid < Pp * 8) sP[tid] = par[(n * Pp) * 8 + tid];
  if (tid < 4) sC[tid] = colors[n * 4 + tid];
  asm volatile("s_wait_asynccnt 0" ::: "memory");
  __syncthreads();

  const int tx = tid & 31;
  const int ty = tid >> 5;
  const int w = blockIdx.x * 32 + tx;
  const int h0 = blockIdx.y * 32 + ty;
  const float fw = (float)w;
  const float cR = sC[0], cG = sC[1], cB = sC[2], cA = sC[3];

  float T[4], aR[4], aG[4], aB[4];
#pragma unroll
  for (int k = 0; k < 4; ++k) {
    T[k] = 1.0f;
    aR[k] = aG[k] = aB[k] = 0.0f;
  }

  // Back-to-front composite: acc += scaled[p]*T_excl ; T *= inv[p]
  for (int pth = Pp - 1; pth >= 0; --pth) {
    const float sx = sP[pth * 8 + 0], sy = sP[pth * 8 + 1];
    const float ct = sP[pth * 8 + 2], st = sP[pth * 8 + 3];
    const float is = sP[pth * 8 + 4], act = sP[pth * 8 + 5];
#pragma unroll
    for (int k = 0; k < 4; ++k) {
      const float fh = (float)(h0 + 8 * k);
      const float dx = fw - sx, dy = fh - sy;
      const float u = (ct * dx + st * dy) * is + 32.5f;   // (BW-1)/2 + 1
      const float v = (-st * dx + ct * dy) * is + 32.5f;  // (BH-1)/2 + 1
      const float u0 = floorf(u), v0 = floorf(v);
      const float du = u - u0, dv = v - v0;
      int u0i = (int)u0;
      u0i = u0i < 0 ? 0 : (u0i > 65 ? 65 : u0i);
      int v0i = (int)v0;
      v0i = v0i < 0 ? 0 : (v0i > 65 ? 65 : v0i);
      int u1i = u0i + 1;
      u1i = u1i > 65 ? 65 : u1i;
      int v1i = v0i + 1;
      v1i = v1i > 65 ? 65 : v1i;

      const int vis[2] = {v0i, v1i};
      const int uis[2] = {u0i, u1i};
      const float wv[2] = {1.0f - dv, dv};
      const float wu[2] = {1.0f - du, du};
      float asamp = 0.0f, msamp = 0.0f;
#pragma unroll
      for (int cv = 0; cv < 2; ++cv) {
#pragma unroll
        for (int cu = 0; cu < 2; ++cu) {
          const int vi = vis[cv], ui = uis[cu];
          const bool inb = (vi >= 1) && (vi <= 64) && (ui >= 1) && (ui <= 64);
          int ri = vi - 1;
          ri = ri < 0 ? 0 : (ri > 63 ? 63 : ri);
          int ci = ui - 1;
          ci = ci < 0 ? 0 : (ci > 63 ? 63 : ci);
          const float wgt = wv[cv] * wu[cu];
          const float bval = sA[ri * 64 + ci];  // LDS gather
          asamp += wgt * (inb ? bval : 0.0f);
          msamp += wgt * (inb ? 1.0f : 0.0f);
        }
      }
      const float alpha = cA * asamp * act;  // inactive => alpha 0 => inv 1
      const float inv = 1.0f - alpha;
      const float at = alpha * T[k];
      aR[k] += (1.0f - cR * msamp) * at;
      aG[k] += (1.0f - cG * msamp) * at;
      aB[k] += (1.0f - cB * msamp) * at;
      T[k] *= inv;
    }
  }

  const long chan = (long)HH * WW;
#pragma unroll
  for (int k = 0; k < 4; ++k) {
    const int h = h0 + 8 * k;
    const long bi = (((long)n * 4) * HH + h) * WW + w;  // images: N x 4 x H x W
    const long bo = (((long)n * 3) * HH + h) * WW + w;  // out:    N x 3 x H x W
    const float i0 = images[bi + 0 * chan];
    const float i1 = images[bi + 1 * chan];
    const float i2 = images[bi + 2 * chan];
    out[bo + 0 * chan] = 1.0f - ((1.0f - i0) * T[k] + aR[k]);
    out[bo + 1 * chan] = 1.0f - ((1.0f - i1) * T[k] + aG[k]);
    out[bo + 2 * chan] = 1.0f - ((1.0f - i2) * T[k] + aB[k]);
  }
}

extern "C" void kernel_launch(void* const* d_in, const int* in_sizes, int n_in,
                              void* d_out, int out_size, void* d_ws,
                              size_t ws_size, hipStream_t stream) {
  (void)in_sizes;
  (void)n_in;
  (void)out_size;
  (void)ws_size;
  const float* images = (const float*)d_in[0];
  const float* knot_t = (const float*)d_in[1];
  const float* a = (const float*)d_in[2];
  const float* b = (const float*)d_in[3];
  const float* two_c = (const float*)d_in[4];
  const float* three_d = (const float*)d_in[5];
  const float* colors = (const float*)d_in[6];
  const float* ts = (const float*)d_in[7];
  const float* brush = (const float*)d_in[8];
  float* out = (float*)d_out;
  float* par = (float*)d_ws;  // N*P*8 floats = 2 KB of scratch

  spline_params_kernel<<<1, 32, 0, stream>>>(knot_t, a, b, two_c, three_d, ts,
                                             par);
  dim3 grid(WW / 32, HH / 32, Nn);
  render_kernel<<<grid, 256, 0, stream>>>(images, colors, brush, par, out);
}